// MPSModel_7224134992119
// MI455X (gfx1250) — compile-verified
//
#include <hip/hip_runtime.h>
#include <hip/hip_bf16.h>

typedef __attribute__((ext_vector_type(2))) float v2f;
typedef __attribute__((ext_vector_type(4))) float v4f;
typedef __attribute__((ext_vector_type(8))) float v8f;

#define F_SITES 512
#define QTR     128     // segment length (4 independent chains of 128 sites)
#define OUTD    8
#define H1D     24

__device__ __forceinline__ float bperm_f32(float v, int srcLane) {
    return __int_as_float(__builtin_amdgcn_ds_bpermute(srcLane << 2, __float_as_int(v)));
}

// v <- v @ M,  M = A0 + x*A1, rows packed: a.x=M[0,0] a.y=M[0,1] a.z=M[1,0] a.w=M[1,1]
__device__ __forceinline__ void vec_stepL(float x, float4 a0, float4 a1, v2f& v) {
    v2f m0 = { fmaf(x, a1.x, a0.x), fmaf(x, a1.y, a0.y) };   // row l=0
    v2f m1 = { fmaf(x, a1.z, a0.z), fmaf(x, a1.w, a0.w) };   // row l=1
    v = v.x * m0 + v.y * m1;
}

// w <- M @ w
__device__ __forceinline__ void vec_stepR(float x, float4 a0, float4 a1, v2f& w) {
    v2f c0 = { fmaf(x, a1.x, a0.x), fmaf(x, a1.z, a0.z) };   // column r=0
    v2f c1 = { fmaf(x, a1.y, a0.y), fmaf(x, a1.w, a0.w) };   // column r=1
    w = c0 * w.x + c1 * w.y;
}

// P <- P @ M  (rows p0, p1)
__device__ __forceinline__ void mat_step(float x, float4 a0, float4 a1, v2f& p0, v2f& p1) {
    v2f m0 = { fmaf(x, a1.x, a0.x), fmaf(x, a1.y, a0.y) };
    v2f m1 = { fmaf(x, a1.z, a0.z), fmaf(x, a1.w, a0.w) };
    p0 = p0.x * m0 + p0.y * m1;
    p1 = p1.x * m0 + p1.y * m1;
}

__global__ __launch_bounds__(128)
void mps_chain_kernel(const float* __restrict__ x,    // [B, 512]
                      const float* __restrict__ A,    // [512, 2, 2, 2]
                      const float* __restrict__ O,    // [2, 8, 2]
                      const float* __restrict__ lb,   // [2]
                      const float* __restrict__ rb,   // [2]
                      const float* __restrict__ W1,   // [24, 8]
                      const float* __restrict__ b1,   // [24]
                      const float* __restrict__ W2,   // [1, 24]
                      const float* __restrict__ b2,   // [1]
                      float* __restrict__ out)        // [B]
{
    __shared__ float4 sA0[F_SITES];   // degree-0 component, rows packed
    __shared__ float4 sA1[F_SITES];   // degree-1 component
    __shared__ float  sOw[4];         // collapsed head: Oeff flattened (l*2+r)
    __shared__ float  sBeff;

    const int tid = threadIdx.x;

    // Repack A[f,l,d,r] -> per-degree row-packed float4 in LDS (16 KB)
    for (int f = tid; f < F_SITES; f += blockDim.x) {
        const float4* a = (const float4*)(A + (size_t)f * 8);
        float4 q0 = a[0];  // A[f,0,0,0..1], A[f,0,1,0..1]
        float4 q1 = a[1];  // A[f,1,0,0..1], A[f,1,1,0..1]
        sA0[f] = make_float4(q0.x, q0.y, q1.x, q1.y);
        sA1[f] = make_float4(q0.z, q0.w, q1.z, q1.w);
    }
    // Collapse the linear head: out = vl^T * Oeff * wr + beff   (exact: head is linear)
    if (tid == 0) {
        float weff[OUTD];
        for (int o = 0; o < OUTD; ++o) {
            float s = 0.f;
            for (int h = 0; h < H1D; ++h) s = fmaf(W2[h], W1[h * OUTD + o], s);
            weff[o] = s;
        }
        float be = b2[0];
        for (int h = 0; h < H1D; ++h) be = fmaf(W2[h], b1[h], be);
        for (int li = 0; li < 2; ++li)
            for (int ri = 0; ri < 2; ++ri) {
                float s = 0.f;
                for (int o = 0; o < OUTD; ++o)
                    s = fmaf(O[li * (OUTD * 2) + o * 2 + ri], weff[o], s);
                sOw[li * 2 + ri] = s;
            }
        sBeff = be;
    }
    __syncthreads();

    const int b = blockIdx.x * blockDim.x + tid;
    const v4f* xrow = (const v4f*)(x + (size_t)b * F_SITES);

    // Four independent dependency chains (reassociated segments):
    //  A: u^T = l^T * M_0..127        (vector scan, ascending)
    //  B: P   = M_128 .. M_255        (matrix product, ascending)
    //  C: Q   = M_256 .. M_383        (matrix product, ascending)
    //  D: t   = M_384..511 * r        (vector scan, descending)
    v2f u  = { lb[0], lb[1] };
    v2f p0 = { 1.f, 0.f }, p1 = { 0.f, 1.f };
    v2f q0 = { 1.f, 0.f }, q1 = { 0.f, 1.f };
    v2f t  = { rb[0], rb[1] };

    for (int j = 0; j < QTR / 4; ++j) {             // 32 iterations, 16 sites each
        v4f xa = __builtin_nontemporal_load(&xrow[j]);           // sites 4j..4j+3
        v4f xb = __builtin_nontemporal_load(&xrow[32 + j]);      // 128+4j ..
        v4f xc = __builtin_nontemporal_load(&xrow[64 + j]);      // 256+4j ..
        v4f xd = __builtin_nontemporal_load(&xrow[127 - j]);     // 508-4j..511-4j
        if (j + 8 < QTR / 4) {
            __builtin_prefetch(&xrow[j + 8], 0, 0);
            __builtin_prefetch(&xrow[32 + j + 8], 0, 0);
            __builtin_prefetch(&xrow[64 + j + 8], 0, 0);
            __builtin_prefetch(&xrow[127 - (j + 8)], 0, 0);
        }
        const int fa = 4 * j;
        const int fb = QTR + 4 * j;
        const int fc = 2 * QTR + 4 * j;
        const int fd = (F_SITES - 4) - 4 * j;

        vec_stepL(xa.x, sA0[fa + 0], sA1[fa + 0], u);
        vec_stepL(xa.y, sA0[fa + 1], sA1[fa + 1], u);
        vec_stepL(xa.z, sA0[fa + 2], sA1[fa + 2], u);
        vec_stepL(xa.w, sA0[fa + 3], sA1[fa + 3], u);

        mat_step(xb.x, sA0[fb + 0], sA1[fb + 0], p0, p1);
        mat_step(xb.y, sA0[fb + 1], sA1[fb + 1], p0, p1);
        mat_step(xb.z, sA0[fb + 2], sA1[fb + 2], p0, p1);
        mat_step(xb.w, sA0[fb + 3], sA1[fb + 3], p0, p1);

        mat_step(xc.x, sA0[fc + 0], sA1[fc + 0], q0, q1);
        mat_step(xc.y, sA0[fc + 1], sA1[fc + 1], q0, q1);
        mat_step(xc.z, sA0[fc + 2], sA1[fc + 2], q0, q1);
        mat_step(xc.w, sA0[fc + 3], sA1[fc + 3], q0, q1);

        vec_stepR(xd.w, sA0[fd + 3], sA1[fd + 3], t);
        vec_stepR(xd.z, sA0[fd + 2], sA1[fd + 2], t);
        vec_stepR(xd.y, sA0[fd + 1], sA1[fd + 1], t);
        vec_stepR(xd.x, sA0[fd + 0], sA1[fd + 0], t);
    }

    // Combine segments: vl^T = u^T P,  wr = Q t
    v2f vl = u.x * p0 + u.y * p1;
    v2f wr = { q0.x * t.x + q0.y * t.y,
               q1.x * t.x + q1.y * t.y };

    // ---- WMMA epilogue: out[b] = z . ow + beff, z = flatten(vl (x) wr) -------
    // Two V_WMMA_F32_16X16X4_F32 per wave (M=16 batch rows each, K=4, B broadcast).
    const float z0 = vl.x * wr.x;
    const float z1 = vl.x * wr.y;
    const float z2 = vl.y * wr.x;
    const float z3 = vl.y * wr.y;

    const int  lane = tid & 31;
    const bool hi   = lane >= 16;

    const float ow0 = sOw[0], ow1 = sOw[1], ow2 = sOw[2], ow3 = sOw[3];
    const float beff = sBeff;

    // A layout (16x4 f32): lanes 0-15 hold {K0,K1} of M=lane; lanes 16-31 hold {K2,K3} of M=lane-16
    float z2m = bperm_f32(z2, lane & 15);          // tile1: K2/K3 of element (lane-16)
    float z3m = bperm_f32(z3, lane & 15);
    float z0p = bperm_f32(z0, (lane & 15) + 16);   // tile2: K0/K1 of element (lane+16)
    float z1p = bperm_f32(z1, (lane & 15) + 16);

    v2f aT1 = { hi ? z2m : z0, hi ? z3m : z1 };    // elements of lanes 0..15
    v2f aT2 = { hi ? z2  : z0p, hi ? z3  : z1p };  // elements of lanes 16..31
    // B layout (4x16 f32): VGPR0 = row K0 (lanes 0-15) / K2 (lanes 16-31); VGPR1 = K1/K3
    v2f bV  = { hi ? ow2 : ow0, hi ? ow3 : ow1 };  // broadcast across N

    v8f cc;
    #pragma unroll
    for (int i = 0; i < 8; ++i) cc[i] = beff;

    v8f d1 = __builtin_amdgcn_wmma_f32_16x16x4_f32(false, aT1, false, bV,
                                                   (short)0, cc, false, false);
    v8f d2 = __builtin_amdgcn_wmma_f32_16x16x4_f32(false, aT2, false, bV,
                                                   (short)0, cc, false, false);

    // All N columns of D are identical (B broadcast), so out[m] sits at static
    // register indices: lanes 0-15 hold D[0..7,*] in d[0..7], lanes 16-31 hold D[8..15,*].
    const int bWave = blockIdx.x * blockDim.x + (tid & ~31);
    if ((lane & 15) == 0) {
        int base = bWave + (hi ? 8 : 0);
        *(float4*)(out + base)     = make_float4(d1[0], d1[1], d1[2], d1[3]);
        *(float4*)(out + base + 4) = make_float4(d1[4], d1[5], d1[6], d1[7]);
    }
    if ((lane & 15) == 1) {
        int base = bWave + 16 + (hi ? 8 : 0);
        *(float4*)(out + base)     = make_float4(d2[0], d2[1], d2[2], d2[3]);
        *(float4*)(out + base + 4) = make_float4(d2[4], d2[5], d2[6], d2[7]);
    }
}

extern "C" void kernel_launch(void* const* d_in, const int* in_sizes, int n_in,
                              void* d_out, int out_size, void* d_ws, size_t ws_size,
                              hipStream_t stream) {
    (void)n_in; (void)d_ws; (void)ws_size; (void)out_size;
    const float* x  = (const float*)d_in[0];
    const float* A  = (const float*)d_in[1];
    const float* O  = (const float*)d_in[2];
    const float* lb = (const float*)d_in[3];
    const float* rb = (const float*)d_in[4];
    const float* W1 = (const float*)d_in[5];
    const float* b1 = (const float*)d_in[6];
    const float* W2 = (const float*)d_in[7];
    const float* b2 = (const float*)d_in[8];
    float* out = (float*)d_out;

    const int B = in_sizes[0] / F_SITES;   // 16384
    const int threads = 128;               // 4 waves/block -> 128 blocks across WGPs
    const int blocks  = B / threads;       // 128 full blocks

    mps_chain_kernel<<<blocks, threads, 0, stream>>>(x, A, O, lb, rb, W1, b1, W2, b2, out);
}